// CombinedCompGCNRepresentations_89163521065629
// MI455X (gfx1250) — compile-verified
//
#include <hip/hip_runtime.h>

#define NUM_ENT   100000
#define NUM_REL   200
#define NUM_EDGE  1000000
#define DIM       128
#define NUM_LAYERS 2
#define BN_EPS    1e-5f

#define XS_STRIDE 132   // 128 + 4 pad -> bank = (row*132+k)%64 varies with row

typedef __attribute__((ext_vector_type(2))) float v2f;
typedef __attribute__((ext_vector_type(8))) float v8f;

__device__ __forceinline__ v8f wmma_f32_k4(v2f a, v2f b, v8f c) {
  // (neg_a, A, neg_b, B, c_mod, C, reuse_a, reuse_b)
  return __builtin_amdgcn_wmma_f32_16x16x4_f32(false, a, false, b, (short)0, c,
                                               false, false);
}

// ---------------------------------------------------------------------------
// Fused 3-way WMMA GEMM over the shared X operand:
//   A  = X @ Wl - sl      (loop term, self-loop folded in)
//   Yf = X @ Wf           (forward message projection)
//   Yb = X @ Wb           (backward message projection)
// Block = 256 threads = 8 waves; block owns a 16(M) x 128(N) tile; the 16x128
// X tile is staged once in LDS (padded stride) and A-fragments are read with
// conflict-free ds_load_b64. Safe to alias Yf with X: each block reads only
// its own 16 rows (behind the barrier) before writing them.
// ---------------------------------------------------------------------------
__global__ void fused_gemm3_wmma(const float* __restrict__ X,
                                 const float* __restrict__ Wl,
                                 const float* __restrict__ Wf,
                                 const float* __restrict__ Wb,
                                 const float* __restrict__ sl,
                                 float* __restrict__ A,
                                 float* __restrict__ Yf,
                                 float* __restrict__ Yb) {
  __shared__ float xs[16 * XS_STRIDE];
  const int m0 = blockIdx.x * 16;

  // cooperative stage: 2048 floats, 2 x float4 per thread, coalesced
#pragma unroll
  for (int j = 0; j < 2; ++j) {
    int idx = threadIdx.x * 4 + j * 1024;
    int r = idx >> 7;
    int cidx = idx & 127;
    float4 v = *(const float4*)(X + (long)(m0 + r) * DIM + cidx);
    *(float4*)(&xs[r * XS_STRIDE + cidx]) = v;
  }
  __syncthreads();

  const int wave = threadIdx.x >> 5;
  const int lane = threadIdx.x & 31;
  const int lr   = lane & 15;
  const int hi   = lane >> 4;
  const int n0   = wave * 16;

  v8f cl = {}, cf = {}, cb = {};
  const float* xrow = &xs[lr * XS_STRIDE + 2 * hi];

#pragma unroll 4
  for (int k0 = 0; k0 < DIM; k0 += 4) {
    v2f a = *(const v2f*)(xrow + k0);           // ds_load_b64, conflict-free
    const int kk = k0 + 2 * hi;
    v2f bl, bf, bb;
    bl[0] = Wl[(long)kk * DIM + n0 + lr];
    bl[1] = Wl[(long)(kk + 1) * DIM + n0 + lr];
    bf[0] = Wf[(long)kk * DIM + n0 + lr];
    bf[1] = Wf[(long)(kk + 1) * DIM + n0 + lr];
    bb[0] = Wb[(long)kk * DIM + n0 + lr];
    bb[1] = Wb[(long)(kk + 1) * DIM + n0 + lr];
    cl = wmma_f32_k4(a, bl, cl);
    cf = wmma_f32_k4(a, bf, cf);
    cb = wmma_f32_k4(a, bb, cb);
  }

  // C/D layout: VGPR r -> row m0 + r + 8*hi, col n0 + lr
  const int col = n0 + lr;
  const float slv = sl[col];
  float* pa = A  + (long)(m0 + 8 * hi) * DIM + col;
  float* pf = Yf + (long)(m0 + 8 * hi) * DIM + col;
  float* pb = Yb + (long)(m0 + 8 * hi) * DIM + col;
#pragma unroll
  for (int r = 0; r < 8; ++r) {
    pa[(long)r * DIM] = cl[r] - slv;
    pf[(long)r * DIM] = cf[r];
    pb[(long)r * DIM] = cb[r];
  }
}

// ---------------------------------------------------------------------------
// Plain WMMA GEMM for the small relation matrices (400 x 128).
// TRANSB=1 computes X @ W^T (nn.Linear).
// ---------------------------------------------------------------------------
template <int TRANSB>
__global__ void gemm_wmma_f32(const float* __restrict__ X,
                              const float* __restrict__ W,
                              float* __restrict__ Y) {
  const int wave = threadIdx.x >> 5;
  const int lane = threadIdx.x & 31;
  const int lr   = lane & 15;
  const int hi   = lane >> 4;
  const int m0   = blockIdx.x * 16;
  const int n0   = wave * 16;

  v8f c = {};
  const float* arow = X + (long)(m0 + lr) * DIM + 2 * hi;

#pragma unroll 4
  for (int k0 = 0; k0 < DIM; k0 += 4) {
    v2f a = *(const v2f*)(arow + k0);
    v2f b;
    const int kk = k0 + 2 * hi;
    if (TRANSB) {
      b[0] = W[(long)(n0 + lr) * DIM + kk];
      b[1] = W[(long)(n0 + lr) * DIM + kk + 1];
    } else {
      b[0] = W[(long)kk * DIM + n0 + lr];
      b[1] = W[(long)(kk + 1) * DIM + n0 + lr];
    }
    c = wmma_f32_k4(a, b, c);
  }

  float* yout = Y + (long)(m0 + 8 * hi) * DIM + n0 + lr;
#pragma unroll
  for (int r = 0; r < 8; ++r) yout[(long)r * DIM] = c[r];
}

// ---------------------------------------------------------------------------
// Degree counting + symmetric edge weights (fwd/bwd share: product symmetric)
// ---------------------------------------------------------------------------
__global__ void count_kernel(const int* __restrict__ idx, unsigned* __restrict__ cnt, int n) {
  int i = blockIdx.x * blockDim.x + threadIdx.x;
  if (i < n) atomicAdd(&cnt[idx[i]], 1u);
}

__global__ void weight_kernel(const int* __restrict__ src, const int* __restrict__ tgt,
                              const unsigned* __restrict__ cs, const unsigned* __restrict__ ct,
                              float* __restrict__ we, int n) {
  int e = blockIdx.x * blockDim.x + threadIdx.x;
  if (e < n) we[e] = rsqrtf((float)cs[src[e]] * (float)ct[tgt[e]]);
}

// sl[d] = self_loop @ w_loop  (1x128 @ 128x128)
__global__ void slvec_kernel(const float* __restrict__ self_loop,
                             const float* __restrict__ w, float* __restrict__ sl) {
  int d = threadIdx.x;
  float s = 0.f;
  for (int k = 0; k < DIM; ++k) s += self_loop[k] * w[k * DIM + d];
  sl[d] = s;
}

// One wave per edge: A[sidx[e]] += (Y[gidx[e]] - Z[2*etype[e]+etoff]) * we[e]
__global__ void edge_scatter_kernel(const float* __restrict__ Y, const float* __restrict__ Z,
                                    const int* __restrict__ gidx, const int* __restrict__ sidx,
                                    const int* __restrict__ etype, int etoff,
                                    const float* __restrict__ we, float* __restrict__ A,
                                    int nedge) {
  long t = (long)blockIdx.x * blockDim.x + threadIdx.x;
  int e = (int)(t >> 5);
  int lane = (int)(t & 31);
  if (e >= nedge) return;
  int g  = gidx[e];
  int s  = sidx[e];
  int zr = 2 * etype[e] + etoff;
  float w = we[e];
  float4 y = ((const float4*)(Y + (long)g * DIM))[lane];
  float4 z = ((const float4*)(Z + (long)zr * DIM))[lane];
  float* out = A + (long)s * DIM + lane * 4;
  atomicAdd(out + 0, (y.x - z.x) * w);
  atomicAdd(out + 1, (y.y - z.y) * w);
  atomicAdd(out + 2, (y.z - z.z) * w);
  atomicAdd(out + 3, (y.w - z.w) * w);
}

// Column sums / sums-of-squares over accum (BatchNorm statistics)
__global__ void colstat_kernel(const float* __restrict__ A, float* __restrict__ S,
                               float* __restrict__ Q, int M) {
  int d  = threadIdx.x;                 // 128 threads = one column each
  int r0 = blockIdx.x * 256;
  int r1 = min(r0 + 256, M);
  float s = 0.f, q = 0.f;
  for (int r = r0; r < r1; ++r) {
    float v = A[(long)r * DIM + d];
    s += v;
    q += v * v;
  }
  atomicAdd(&S[d], s);
  atomicAdd(&Q[d], q);
}

// BatchNorm (bias cancels inside x-mean; /3 folds into the scale):
// out = (a - mean_a)/3 * rsqrt(var_a/9 + eps) * gamma + beta
__global__ void bn_kernel(const float* __restrict__ A, const float* __restrict__ S,
                          const float* __restrict__ Q, const float* __restrict__ gamma,
                          const float* __restrict__ beta, float* __restrict__ out, int M) {
  int i = blockIdx.x * blockDim.x + threadIdx.x;
  if (i >= M * DIM) return;
  int d = i & (DIM - 1);
  float invM  = 1.0f / (float)M;
  float mean  = S[d] * invM;
  float var   = Q[d] * invM - mean * mean;
  float scale = rsqrtf(var * (1.0f / 9.0f) + BN_EPS) * (1.0f / 3.0f);
  out[i] = (A[i] - mean) * scale * gamma[d] + beta[d];
}

// ---------------------------------------------------------------------------
extern "C" void kernel_launch(void* const* d_in, const int* in_sizes, int n_in,
                              void* d_out, int out_size, void* d_ws, size_t ws_size,
                              hipStream_t stream) {
  const float* x_e      = (const float*)d_in[0];
  const float* x_r      = (const float*)d_in[1];
  const int*   eidx     = (const int*)d_in[2];   // [2, E]
  const int*   etype    = (const int*)d_in[3];
  const float* w_loop   = (const float*)d_in[4]; // [L,128,128]
  const float* w_fwd    = (const float*)d_in[5];
  const float* w_bwd    = (const float*)d_in[6];
  const float* w_rel    = (const float*)d_in[7];
  const float* selfloop = (const float*)d_in[8]; // [L,1,128]
  const float* gamma    = (const float*)d_in[10];
  const float* beta     = (const float*)d_in[11];

  const int* src = eidx;
  const int* tgt = eidx + NUM_EDGE;

  // ---- workspace carve (floats) ----
  float* ws = (float*)d_ws;
  size_t off = 0;
  float*    we    = ws + off; off += NUM_EDGE;
  unsigned* cnt_s = (unsigned*)(ws + off); off += NUM_ENT;
  unsigned* cnt_t = (unsigned*)(ws + off); off += NUM_ENT;
  float* A   = ws + off; off += (size_t)NUM_ENT * DIM;   // message accumulator
  float* Yf  = ws + off; off += (size_t)NUM_ENT * DIM;   // fwd proj / next-layer X
  float* Yb  = ws + off; off += (size_t)NUM_ENT * DIM;   // bwd proj
  float* Zf  = ws + off; off += (size_t)2 * NUM_REL * DIM;
  float* Zb  = ws + off; off += (size_t)2 * NUM_REL * DIM;
  float* Rn  = ws + off; off += (size_t)2 * NUM_REL * DIM;
  float* sl  = ws + off; off += DIM;
  float* S   = ws + off; off += DIM;
  float* Q   = ws + off; off += DIM;
  (void)ws_size; (void)n_in; (void)in_sizes; (void)out_size;

  const int TOT   = NUM_ENT * DIM;          // 12.8M
  const int BLK   = 256;
  const int gTot  = TOT / BLK;              // exact
  const int gEdge = (NUM_EDGE + BLK - 1) / BLK;
  const int gEdgeWave = (int)(((long)NUM_EDGE * 32 + BLK - 1) / BLK);
  const int gEnt16 = NUM_ENT / 16;          // 6250 (exact)
  const int gRel16 = (2 * NUM_REL) / 16;    // 25 (exact)
  const int gStat  = (NUM_ENT + 255) / 256;

  // ---- edge weights (once; symmetric for both directions) ----
  hipMemsetAsync(cnt_s, 0, 2 * (size_t)NUM_ENT * sizeof(unsigned), stream);
  count_kernel<<<gEdge, BLK, 0, stream>>>(src, cnt_s, NUM_EDGE);
  count_kernel<<<gEdge, BLK, 0, stream>>>(tgt, cnt_t, NUM_EDGE);
  weight_kernel<<<gEdge, BLK, 0, stream>>>(src, tgt, cnt_s, cnt_t, we, NUM_EDGE);

  const float* Xin = x_e;
  const float* Rin = x_r;
  for (int l = 0; l < NUM_LAYERS; ++l) {
    const float* wl = w_loop + (size_t)l * DIM * DIM;
    const float* wf = w_fwd  + (size_t)l * DIM * DIM;
    const float* wb = w_bwd  + (size_t)l * DIM * DIM;
    const float* wr = w_rel  + (size_t)l * DIM * DIM;

    // self-loop row vector, then fused 3-projection pass over X:
    //   A = X@wl - sl,  Yf = X@wf,  Yb = X@wb   (single read of X)
    slvec_kernel<<<1, DIM, 0, stream>>>(selfloop + (size_t)l * DIM, wl, sl);
    fused_gemm3_wmma<<<gEnt16, BLK, 0, stream>>>(Xin, wl, wf, wb, sl, A, Yf, Yb);

    // relation projections (tiny)
    gemm_wmma_f32<0><<<gRel16, BLK, 0, stream>>>(Rin, wf, Zf);
    gemm_wmma_f32<0><<<gRel16, BLK, 0, stream>>>(Rin, wb, Zb);

    // message scatter (bandwidth-bound; f32 atomics land in L2)
    edge_scatter_kernel<<<gEdgeWave, BLK, 0, stream>>>(Yf, Zf, src, tgt, etype, 0,
                                                       we, A, NUM_EDGE);
    edge_scatter_kernel<<<gEdgeWave, BLK, 0, stream>>>(Yb, Zb, tgt, src, etype, 1,
                                                       we, A, NUM_EDGE);

    // BatchNorm over entities; output aliases Yf (consumed above) or d_out
    hipMemsetAsync(S, 0, 2 * DIM * sizeof(float), stream);  // S,Q contiguous
    colstat_kernel<<<gStat, DIM, 0, stream>>>(A, S, Q, NUM_ENT);
    float* Xout = (l == NUM_LAYERS - 1) ? (float*)d_out : Yf;
    bn_kernel<<<gTot, BLK, 0, stream>>>(A, S, Q, gamma + (size_t)l * DIM,
                                        beta + (size_t)l * DIM, Xout, NUM_ENT);

    // relation update: x_r_new = x_r @ w_rel^T
    float* Rout = (l == NUM_LAYERS - 1) ? ((float*)d_out + (size_t)NUM_ENT * DIM) : Rn;
    gemm_wmma_f32<1><<<gRel16, BLK, 0, stream>>>(Rin, wr, Rout);

    Xin = Xout;
    Rin = Rout;
  }
}